// NeighborSearch_22411139350816
// MI455X (gfx1250) — compile-verified
//
#include <hip/hip_runtime.h>

// CDNA5 / gfx1250, wave32. One V_WMMA_F32_16X16X4_F32 per 16x16 distance tile.
typedef __attribute__((ext_vector_type(2))) float v2f;
typedef __attribute__((ext_vector_type(8))) float v8f;

#define NS_EPS 1e-7f
#define NS_MAX_NBRS 262144

// Fast hardware sqrt (single v_sqrt_f32, ~1 ulp) -- avoids hipcc's IEEE
// refinement sequence (~15 VALU ops/elem) which would make the 144M-element
// epilogue VALU-bound instead of HBM-store-bound.
__device__ __forceinline__ float ns_fast_sqrtf(float x) {
  return __builtin_amdgcn_sqrtf(x);
}

// ---------------------------------------------------------------------------
// Kernel 1: precompute squared norms, zero counts + neighbor-index region.
// ---------------------------------------------------------------------------
__global__ void ns_init(const float* __restrict__ data,
                        const float* __restrict__ queries,
                        float* __restrict__ d2, float* __restrict__ q2,
                        unsigned* __restrict__ counts,
                        int* __restrict__ nbrs,
                        int n, int m, int maxNbrs) {
  int i = blockIdx.x * blockDim.x + threadIdx.x;
  if (i < n) {
    float x = data[3 * i + 0], y = data[3 * i + 1], z = data[3 * i + 2];
    d2[i] = x * x + y * y + z * z;
  }
  if (i < m) {
    float x = queries[3 * i + 0], y = queries[3 * i + 1], z = queries[3 * i + 2];
    q2[i] = x * x + y * y + z * z;
    counts[i] = 0u;
  }
  if (i < maxNbrs) nbrs[i] = 0;
}

// ---------------------------------------------------------------------------
// Kernel 2: WMMA tile kernel. One wave (32 lanes) owns one 16x16 tile of the
// [m, n] distance matrix. K=3 padded to 4 to match V_WMMA_F32_16X16X4_F32.
//
// A (16x4, 32-bit): lanes 0-15 hold row M=lane, VGPR0=K0, VGPR1=K1;
//                   lanes 16-31 hold row M=lane-16, VGPR0=K2, VGPR1=K3(=0).
// B (4x16): mirrored (column N = lane&15, same K split).
// C/D (16x16 f32): VGPR r -> M=r (lanes 0-15) / M=r+8 (lanes 16-31), N=lane&15.
// ---------------------------------------------------------------------------
__global__ __launch_bounds__(256) void ns_dists_wmma(
    const float* __restrict__ data, const float* __restrict__ queries,
    const float* __restrict__ radiusPtr,
    const float* __restrict__ d2, const float* __restrict__ q2,
    float* __restrict__ dists, unsigned* __restrict__ counts,
    int m, int n, int tilesD, long long totalTiles) {
  const int lane = threadIdx.x & 31;
  const int wid = threadIdx.x >> 5;
  const long long tile = (long long)blockIdx.x * (blockDim.x >> 5) + wid;
  if (tile >= totalTiles) return;  // wave-uniform: EXEC stays all-1s below

  const int tileQ = (int)(tile / tilesD);
  const int tileDd = (int)(tile % tilesD);
  const int qBase = tileQ * 16;
  const int dBase = tileDd * 16;

  const int l15 = lane & 15;
  const int half = lane >> 4;  // 0: K=0,1  1: K=2,3

  // Clamped loads keep EXEC full (no divergence before the WMMA).
  const int qrow = min(qBase + l15, m - 1);
  const int dcol = min(dBase + l15, n - 1);
  const float qx = queries[3 * qrow + 0];
  const float qy = queries[3 * qrow + 1];
  const float qz = queries[3 * qrow + 2];
  const float dx = data[3 * dcol + 0];
  const float dy = data[3 * dcol + 1];
  const float dz = data[3 * dcol + 2];

  v2f a, b;
  a.x = half ? qz : qx;
  a.y = half ? 0.0f : qy;
  b.x = half ? dz : dx;
  b.y = half ? 0.0f : dy;

  v8f c = {};
  // D[mrow][ncol] = q_mrow . d_ncol
  c = __builtin_amdgcn_wmma_f32_16x16x4_f32(
      /*neg_a=*/false, a, /*neg_b=*/false, b,
      /*c_mod=*/(short)0, c, /*reuse_a=*/false, /*reuse_b=*/false);

  const float R = radiusPtr[0];
  const float d2n = d2[dcol];
  const int mOff = half * 8;
  const int ncol = dBase + l15;

  const bool interior = (qBase + 16 <= m) && (dBase + 16 <= n);  // wave-uniform
  if (interior) {
    // Fast path: no per-element bounds checks, ballot-aggregated counting
    // (atomics only fire for (tile,row) pairs that actually have hits).
    // The 8 row-norms per lane are contiguous and 32B-aligned: two b128 loads
    // instead of eight b32 loads.
    const float4 qa = *(const float4*)(q2 + qBase + mOff);
    const float4 qb = *(const float4*)(q2 + qBase + mOff + 4);
    const float q2v[8] = {qa.x, qa.y, qa.z, qa.w, qb.x, qb.y, qb.z, qb.w};

    float* rowPtr = dists + (size_t)(qBase + mOff) * (size_t)n + (size_t)ncol;
#pragma unroll
    for (int r = 0; r < 8; ++r) {
      const float sq = q2v[r] + d2n - 2.0f * c[r];
      float dist = ns_fast_sqrtf(fmaxf(sq, 0.0f));
      dist = (dist == 0.0f) ? NS_EPS : dist;
      const float outv = (dist <= R) ? dist : 0.0f;
      rowPtr[(size_t)r * (size_t)n] = outv;

      const unsigned mask = (unsigned)__ballot(outv > 0.0f);
      const unsigned lo = mask & 0xFFFFu;
      const unsigned hi = mask >> 16;
      if (lane == 0 && lo) atomicAdd(&counts[qBase + r], (unsigned)__popc(lo));
      if (lane == 16 && hi)
        atomicAdd(&counts[qBase + r + 8], (unsigned)__popc(hi));
    }
  } else {
    // Edge tiles (unused when m,n are multiples of 16): guarded generic path.
#pragma unroll
    for (int r = 0; r < 8; ++r) {
      const int mrow = qBase + r + mOff;
      if (mrow < m && ncol < n) {
        const float sq = q2[mrow] + d2n - 2.0f * c[r];
        float dist = ns_fast_sqrtf(fmaxf(sq, 0.0f));
        dist = (dist == 0.0f) ? NS_EPS : dist;
        const float outv = (dist <= R) ? dist : 0.0f;
        dists[(size_t)mrow * (size_t)n + (size_t)ncol] = outv;
        if (outv > 0.0f) atomicAdd(&counts[mrow], 1u);
      }
    }
  }
}

// ---------------------------------------------------------------------------
// Kernel 3: single-block exclusive scan: counts[m] -> splits[m+1] (int32).
// ---------------------------------------------------------------------------
__global__ __launch_bounds__(256) void ns_scan(const unsigned* __restrict__ counts,
                                               int* __restrict__ splits, int m) {
  __shared__ unsigned partial[256];
  const int t = threadIdx.x;
  const int chunk = (m + 255) / 256;
  const int start = t * chunk;
  const int end = min(start + chunk, m);

  unsigned s = 0;
  for (int i = start; i < end; ++i) s += counts[i];
  partial[t] = s;
  __syncthreads();

  if (t == 0) {
    unsigned run = 0;
    for (int i = 0; i < 256; ++i) {
      const unsigned v = partial[i];
      partial[i] = run;
      run += v;
    }
  }
  __syncthreads();

  unsigned run = partial[t];
  if (t == 0) splits[0] = 0;
  for (int i = start; i < end; ++i) {
    run += counts[i];
    splits[i + 1] = (int)run;  // inclusive prefix -> splits[i+1]
  }
}

// ---------------------------------------------------------------------------
// Kernel 4: one wave per row; ballot/popcount compaction of in-radius column
// indices in ascending order (matches row-major jnp.nonzero ordering).
// ---------------------------------------------------------------------------
__global__ __launch_bounds__(256) void ns_compact(
    const float* __restrict__ dists, const int* __restrict__ splits,
    int* __restrict__ nbrs, int mRows, int nCols, int maxNbrs) {
  const int wavesPerBlock = blockDim.x >> 5;
  const int wid = threadIdx.x >> 5;
  const int lane = threadIdx.x & 31;
  const int row = blockIdx.x * wavesPerBlock + wid;
  if (row >= mRows) return;

  unsigned running = (unsigned)splits[row];
  const float* drow = dists + (size_t)row * (size_t)nCols;

  for (int n0 = 0; n0 < nCols; n0 += 32) {
    const int nc = n0 + lane;
    const bool pred = (nc < nCols) && (drow[nc] > 0.0f);
    const unsigned long long mask = __ballot(pred);
    const unsigned pos = (unsigned)__popcll(mask & ((1ull << lane) - 1ull));
    const unsigned idx = running + pos;
    if (pred && idx < (unsigned)maxNbrs) nbrs[idx] = nc;
    running += (unsigned)__popcll(mask);
  }
}

// ---------------------------------------------------------------------------
// Host launcher.
// d_out layout (flat, return order): dists [m*n] f32 | nbrs [262144] i32 |
// splits [m+1] i32.  d_ws layout: d2 [n] f32 | q2 [m] f32 | counts [m] u32.
// ---------------------------------------------------------------------------
extern "C" void kernel_launch(void* const* d_in, const int* in_sizes, int n_in,
                              void* d_out, int out_size, void* d_ws, size_t ws_size,
                              hipStream_t stream) {
  const float* data = (const float*)d_in[0];     // [n, 3]
  const float* queries = (const float*)d_in[1];  // [m, 3]
  const float* radius = (const float*)d_in[2];   // scalar

  const int n = in_sizes[0] / 3;
  const int m = in_sizes[1] / 3;

  float* dists = (float*)d_out;
  int* nbrs = (int*)(dists + (size_t)m * (size_t)n);
  int* splits = nbrs + NS_MAX_NBRS;

  float* d2 = (float*)d_ws;
  float* q2 = d2 + n;
  unsigned* counts = (unsigned*)(q2 + m);

  // 1) init: norms, zero counts, zero neighbor-index region
  int initN = NS_MAX_NBRS;
  if (m > initN) initN = m;
  if (n > initN) initN = n;
  ns_init<<<(initN + 255) / 256, 256, 0, stream>>>(data, queries, d2, q2,
                                                   counts, nbrs, n, m,
                                                   NS_MAX_NBRS);

  // 2) WMMA distance tiles: one wave per 16x16 tile, 8 waves / 256-thread block
  const int tilesQ = (m + 15) / 16;
  const int tilesD = (n + 15) / 16;
  const long long totalTiles = (long long)tilesQ * (long long)tilesD;
  const int wavesPerBlock = 8;
  const int blocks = (int)((totalTiles + wavesPerBlock - 1) / wavesPerBlock);
  ns_dists_wmma<<<blocks, 256, 0, stream>>>(data, queries, radius, d2, q2,
                                            dists, counts, m, n, tilesD,
                                            totalTiles);

  // 3) splits = exclusive scan of counts
  ns_scan<<<1, 256, 0, stream>>>(counts, splits, m);

  // 4) ordered CSR compaction, one wave per row
  const int rowsPerBlock = 8;
  ns_compact<<<(m + rowsPerBlock - 1) / rowsPerBlock, 32 * rowsPerBlock, 0,
               stream>>>(dists, splits, nbrs, m, n, NS_MAX_NBRS);
}